// VolumeRenderer_57612691308835
// MI455X (gfx1250) — compile-verified
//
#include <hip/hip_runtime.h>
#include <stdint.h>

// Volume rendering (NeRF-style ray marching) for MI455X / gfx1250.
// Bandwidth-bound: ~322 MiB traffic -> ~14 us floor at 23.3 TB/s.
// Uses CDNA5 async global->LDS DMA (ASYNCcnt) with per-lane self-staging,
// double-buffered chunks, no workgroup barriers.

#define S_SAMPLES 128
#define CH 8                    // samples per staged chunk
#define NCH (S_SAMPLES / CH)    // 16 chunks
#define BLOCK 128               // threads per block (4 waves), 1 thread = 1 ray
#define DROW 12                 // 8 floats data + 4 pad  (48 B, 16B-aligned rows)
#define FROW 28                 // 24 floats data + 4 pad (112 B, 16B-aligned rows)
#define ASYNC_PER_CHUNK 10      // 2 (density) + 2 (depth) + 6 (feature) b128 ops
#define FAR_DELTA 1e10f

// Low 32 bits of a generic pointer to LDS == byte offset within the
// workgroup's LDS allocation (ISA: flat->LDS uses addr[31:0]; async VDST is
// relative to LDS_BASE).
__device__ __forceinline__ uint32_t lds_off_of(const void* p) {
    return (uint32_t)(uintptr_t)p;
}

// CDNA5 async DMA: copy 16 bytes per lane from global memory to LDS.
// Tracked by ASYNCcnt (not LOADcnt).
__device__ __forceinline__ void async_ld_b128(uint32_t lds, const void* gaddr) {
    asm volatile("global_load_async_to_lds_b128 %0, %1, off"
                 :: "v"(lds), "v"(gaddr)
                 : "memory");
}

// Split-wait on the wave's ASYNCcnt. Async loads retire in order, so
// waiting <= ASYNC_PER_CHUNK guarantees the older chunk's LDS writes landed
// while the prefetched chunk stays in flight.
#define WAIT_ASYNC(n) asm volatile("s_wait_asynccnt " #n ::: "memory")

__global__ __launch_bounds__(BLOCK)
void VolumeRenderer_57612691308835_kernel(const float* __restrict__ density,
                                          const float* __restrict__ feature,
                                          const float* __restrict__ depth,
                                          float* __restrict__ out_feat,   // [N,3]
                                          float* __restrict__ out_depth,  // [N]
                                          int n_rays) {
    __shared__ __align__(16) float s_den[2][BLOCK][DROW];
    __shared__ __align__(16) float s_dep[2][BLOCK][DROW];
    __shared__ __align__(16) float s_fea[2][BLOCK][FROW];

    const int t = threadIdx.x;
    const int r = blockIdx.x * BLOCK + t;
    if (r >= n_rays) return;

    const float* gD = density + (size_t)r * S_SAMPLES;
    const float* gZ = depth   + (size_t)r * S_SAMPLES;
    const float* gF = feature + (size_t)r * S_SAMPLES * 3;

    uint32_t lD[2], lZ[2], lF[2];
    lD[0] = lds_off_of(&s_den[0][t][0]); lD[1] = lds_off_of(&s_den[1][t][0]);
    lZ[0] = lds_off_of(&s_dep[0][t][0]); lZ[1] = lds_off_of(&s_dep[1][t][0]);
    lF[0] = lds_off_of(&s_fea[0][t][0]); lF[1] = lds_off_of(&s_fea[1][t][0]);

    // Each lane stages only the data it will consume -> no barriers needed.
    auto issue_chunk = [&](int k, int b) {
        const char* pd = (const char*)(gD + k * CH);
        const char* pz = (const char*)(gZ + k * CH);
        const char* pf = (const char*)(gF + k * CH * 3);
#pragma unroll
        for (int j = 0; j < 2; ++j)  // 32 B density
            async_ld_b128(lD[b] + 16u * j, pd + 16 * j);
#pragma unroll
        for (int j = 0; j < 2; ++j)  // 32 B depth
            async_ld_b128(lZ[b] + 16u * j, pz + 16 * j);
#pragma unroll
        for (int j = 0; j < 6; ++j)  // 96 B feature
            async_ld_b128(lF[b] + 16u * j, pf + 16 * j);
    };

    issue_chunk(0, 0);

    float T = 1.0f;
    float a0 = 0.0f, a1 = 0.0f, a2 = 0.0f, az = 0.0f;
    // Carried previous-sample state (handles delta[i] = z[i+1] - z[i]).
    float ps = 0.0f, pz = 0.0f, pf0 = 0.0f, pf1 = 0.0f, pf2 = 0.0f;

    for (int k = 0; k < NCH; ++k) {
        const int b = k & 1;
        if (k + 1 < NCH) {
            issue_chunk(k + 1, (k + 1) & 1);  // prefetch into the other buffer
            WAIT_ASYNC(10);                   // chunk k landed; k+1 in flight
        } else {
            WAIT_ASYNC(0);                    // drain everything for last chunk
        }

#pragma unroll
        for (int s = 0; s < CH; ++s) {
            const float sg = s_den[b][t][s];
            const float z  = s_dep[b][t][s];
            const float f0 = s_fea[b][t][3 * s + 0];
            const float f1 = s_fea[b][t][3 * s + 1];
            const float f2 = s_fea[b][t][3 * s + 2];
            if ((k != 0) || (s != 0)) {
                // Process sample (global index - 1) now that we have its delta.
                const float delta = z - pz;
                const float att   = __expf(-ps * delta);
                T *= att;                         // inclusive prefix product
                const float w = T * (1.0f - att); // weight
                a0 += w * pf0;
                a1 += w * pf1;
                a2 += w * pf2;
                az += w * pz;
            }
            ps = sg; pz = z; pf0 = f0; pf1 = f1; pf2 = f2;
        }
    }

    // Last sample uses the FAR_DELTA sentinel.
    {
        const float att = __expf(-ps * FAR_DELTA);  // -> 0 for sigma > 0, 1 at sigma == 0
        T *= att;
        const float w = T * (1.0f - att);
        a0 += w * pf0;
        a1 += w * pf1;
        a2 += w * pf2;
        az += w * pz;
    }

    out_feat[3 * r + 0] = a0;
    out_feat[3 * r + 1] = a1;
    out_feat[3 * r + 2] = a2;
    out_depth[r] = az;
}

extern "C" void kernel_launch(void* const* d_in, const int* in_sizes, int n_in,
                              void* d_out, int out_size, void* d_ws, size_t ws_size,
                              hipStream_t stream) {
    // setup_inputs() order: density [N,S], feature [N,S,3], depth_values [N,S]
    const float* density = (const float*)d_in[0];
    const float* feature = (const float*)d_in[1];
    const float* depth   = (const float*)d_in[2];

    const int n_rays = in_sizes[0] / S_SAMPLES;

    // Outputs concatenated flat in return order: feat_out [N,3] then depth_out [N].
    float* out_feat  = (float*)d_out;
    float* out_depth = out_feat + (size_t)n_rays * 3;

    const int grid = (n_rays + BLOCK - 1) / BLOCK;
    VolumeRenderer_57612691308835_kernel<<<grid, BLOCK, 0, stream>>>(
        density, feature, depth, out_feat, out_depth, n_rays);
}